// VectorQuantizer_2594160246935
// MI455X (gfx1250) — compile-verified
//
#include <hip/hip_runtime.h>
#include <hip/hip_bf16.h>
#include <math.h>

// Problem constants (from reference)
#define BB   16
#define EE   64
#define TT   8192
#define KCB  2048
#define NROW (BB * TT)            // 131072 flattened rows
#define OUT_ELEMS ((size_t)BB * EE * TT)  // 8388608
#define ROWS_PER_BLOCK 128
#define NBLK (NROW / ROWS_PER_BLOCK)      // 1024
#define CHUNK_ENT 64              // codebook entries staged in LDS per iteration
#define NCHUNK (KCB / CHUNK_ENT)  // 32
#define LDS_STRIDE 68             // 64 + 4 pad -> conflict-free, matches TDM pad
#define BUF_FLOATS (CHUNK_ENT * LDS_STRIDE)

typedef float v2f __attribute__((ext_vector_type(2)));
typedef float v8f __attribute__((ext_vector_type(8)));
typedef unsigned int u32;
typedef u32 u32x4 __attribute__((ext_vector_type(4)));
typedef int  i32x4 __attribute__((ext_vector_type(4)));
typedef int  i32x8 __attribute__((ext_vector_type(8)));

// ---------------------------------------------------------------------------
// TDM: async DMA of one codebook chunk (64 entries x 64 f32) global -> LDS.
// Hardware padding: +4 DWORDs after every 64 DWORDs => LDS row stride 68
// floats, exactly the bank-conflict-free layout the compute loop reads.
// ---------------------------------------------------------------------------
__device__ __forceinline__ void tdm_load_cb_chunk(const float* gsrc, float* lds_dst) {
  const u32 lds_off = (u32)(unsigned long long)(uintptr_t)lds_dst; // flat LDS addr[31:0] = byte offset
  const unsigned long long ga = (unsigned long long)(uintptr_t)gsrc;

  u32x4 g0;
  g0[0] = 1u;                                        // count=1 (valid user D#)
  g0[1] = lds_off;                                   // lds_addr (bytes)
  g0[2] = (u32)(ga & 0xFFFFFFFFull);                 // global_addr[31:0]
  g0[3] = (u32)((ga >> 32) & 0x01FFFFFFull)          // global_addr[56:32]
        | (2u << 30);                                // type = 2 ("image")

  i32x8 g1;
  g1[0] = (int)((2u << 16)      // data_size = 2 -> 4-byte elements
              | (1u << 20)      // pad_enable
              | (5u << 22)      // pad_interval = 5 -> every 64 DWORDs
              | (3u << 25));    // pad_amount  = 3 -> +4 DWORDs
  g1[1] = (int)(64u << 16);     // tensor_dim0[15:0] = 64 (atomic_barrier_addr=0)
  g1[2] = (int)(2048u << 16);   // tensor_dim0 hi = 0 | tensor_dim1[15:0] = 2048
  g1[3] = (int)(64u << 16);     // tensor_dim1 hi = 0 | tile_dim0 = 64
  g1[4] = (int)(64u);           // tile_dim1 = 64 entries | tile_dim2 = 0
  g1[5] = 64;                   // tensor_dim0_stride[31:0] = 64 elements
  g1[6] = 0;                    // stride hi = 0 | tensor_dim1_stride lo = 0
  g1[7] = 0;

  i32x4 z4 = {0, 0, 0, 0};
#if __clang_major__ >= 23
  i32x8 z8 = {0, 0, 0, 0, 0, 0, 0, 0};
  __builtin_amdgcn_tensor_load_to_lds(g0, g1, z4, z4, z8, 0);
#else
  __builtin_amdgcn_tensor_load_to_lds(g0, g1, z4, z4, 0);
#endif
}

// ---------------------------------------------------------------------------
// Kernel 1: codebook squared norms  cb2[n] = sum_e codebook[n][e]^2
// ---------------------------------------------------------------------------
__global__ __launch_bounds__(256) void vq_cb2_kernel(const float* __restrict__ cb,
                                                     float* __restrict__ cb2) {
  int n = blockIdx.x * 256 + threadIdx.x;   // grid sized exactly: 8*256 = 2048
  const float4* p = (const float4*)(cb + (size_t)n * EE);
  float s = 0.f;
#pragma unroll
  for (int i = 0; i < EE / 4; ++i) {
    float4 q = p[i];
    s += q.x * q.x + q.y * q.y + q.z * q.z + q.w * q.w;
  }
  cb2[n] = s;
}

// ---------------------------------------------------------------------------
// Kernel 2: main VQ kernel.
// One wave = 16 rows. Block = 8 waves = 128 rows (consecutive t within one b).
// Codebook streamed by TDM (double-buffered, overlapped with compute).
// Per 16-entry tile: 16x WMMA_F32_16X16X4_F32 accumulate dot = xf * cbT,
// then running min/argmin of (cb2[n] - 2*dot).
// ---------------------------------------------------------------------------
__global__ __launch_bounds__(256) void vq_main_kernel(const float* __restrict__ x,
                                                      const float* __restrict__ cb,
                                                      const float* __restrict__ cb2,
                                                      float* __restrict__ out,
                                                      float* __restrict__ pfit,
                                                      float* __restrict__ psx,
                                                      float* __restrict__ psx2) {
  __shared__ float lds_cb[2 * BUF_FLOATS];          // double buffer, 34816 B
  __shared__ int   best_lds[ROWS_PER_BLOCK];
  __shared__ float wfit[8], wsx[8], wsx2[8];

  const int tid  = threadIdx.x;
  const int lane = tid & 31;
  const int wave = tid >> 5;
  const int mm   = lane & 15;      // row (for A) / column-entry (for B/D)
  const int hf   = lane >> 4;      // lane half: selects K sub-pair

  const int n0   = blockIdx.x * ROWS_PER_BLOCK;    // first flattened row
  const int bblk = n0 >> 13;                       // T = 8192 = 2^13
  const int t0   = n0 & (TT - 1);

  // ---- Kick off TDM for chunk 0 while we load A fragments ----
  if (wave == 0) tdm_load_cb_chunk(cb, &lds_cb[0]);

  // ---- Load A fragments once: a[s][j] = xf[row][e = 4s + 2*hf + j] ----
  const int   nrow = n0 + wave * 16 + mm;
  const int   trow = nrow & (TT - 1);
  const float* px  = x + (size_t)bblk * (EE * TT) + trow;

  float a[16][2];
  float sx = 0.f, sx2 = 0.f;   // per-lane partial sums of x and x^2
#pragma unroll
  for (int s = 0; s < 16; ++s) {
#pragma unroll
    for (int j = 0; j < 2; ++j) {
      float v = px[(size_t)((s << 2) + (hf << 1) + j) * TT];
      a[s][j] = v;
      sx  += v;
      sx2 += v * v;
    }
  }

  // Running argmin state: slot v covers D row (v + 8*hf), column mm.
  float minv[8];
  int   mini[8];
#pragma unroll
  for (int v = 0; v < 8; ++v) { minv[v] = 3.4e38f; mini[v] = 0; }

  for (int chunk = 0; chunk < NCHUNK; ++chunk) {   // 32 chunks
    const int cb0 = chunk * CHUNK_ENT;

    // Issue next chunk's DMA, then retire the current one (in-order TENSORcnt)
    if (wave == 0) {
      if (chunk + 1 < NCHUNK) {
        tdm_load_cb_chunk(cb + (size_t)(cb0 + CHUNK_ENT) * EE,
                          &lds_cb[((chunk + 1) & 1) * BUF_FLOATS]);
        __builtin_amdgcn_s_wait_tensorcnt(1);   // chunk's DMA complete
      } else {
        __builtin_amdgcn_s_wait_tensorcnt(0);
      }
    }
    __syncthreads();   // chunk data visible to all 8 waves

    const float* buf = &lds_cb[(chunk & 1) * BUF_FLOATS];

#pragma unroll
    for (int ct = 0; ct < 4; ++ct) {         // four 16-entry tiles per chunk
      const int   ent0 = ct << 4;
      const float cb2v = cb2[cb0 + ent0 + mm];
      const float* bp  = &buf[(ent0 + mm) * LDS_STRIDE + (hf << 1)];

      v8f acc = {0.f, 0.f, 0.f, 0.f, 0.f, 0.f, 0.f, 0.f};
#pragma unroll
      for (int s = 0; s < 16; ++s) {         // K = 64 in steps of 4
        v2f av; av[0] = a[s][0]; av[1] = a[s][1];
        v2f bv = *(const v2f*)(bp + (s << 2));   // 8B-aligned ds_load
        acc = __builtin_amdgcn_wmma_f32_16x16x4_f32(
            /*neg_a=*/false, av, /*neg_b=*/false, bv,
            /*c_mod=*/(short)0, acc, /*reuse_a=*/false, /*reuse_b=*/false);
      }

      const int nidx = cb0 + ent0 + mm;      // codebook index of this column
#pragma unroll
      for (int v = 0; v < 8; ++v) {
        float cand = fmaf(-2.0f, acc[v], cb2v);   // cb2 - 2*dot
        if (cand < minv[v] || (cand == minv[v] && nidx < mini[v])) {
          minv[v] = cand; mini[v] = nidx;
        }
      }
    }
    __syncthreads();   // all waves done with this buffer before TDM reuses it
  }

  // ---- Cross-lane argmin within each 16-lane half (columns 0..15) ----
#pragma unroll
  for (int v = 0; v < 8; ++v) {
#pragma unroll
    for (int m = 1; m <= 8; m <<= 1) {
      float ov = __shfl_xor(minv[v], m, 32);
      int   oi = __shfl_xor(mini[v], m, 32);
      if (ov < minv[v] || (ov == minv[v] && oi < mini[v])) {
        minv[v] = ov; mini[v] = oi;
      }
    }
  }

  // fit per wave = sum_lanes sx2 (= sum_rows ||x||^2) + sum_rows min(cb2-2dot)
  float fitl = sx2;
  if (mm == 0) {
#pragma unroll
    for (int v = 0; v < 8; ++v) {
      fitl += minv[v];
      best_lds[wave * 16 + (hf << 3) + v] = mini[v];  // row = v + 8*hf
    }
  }

  float rsx = sx, rsx2 = sx2;
#pragma unroll
  for (int m = 1; m <= 16; m <<= 1) {
    fitl += __shfl_xor(fitl, m, 32);
    rsx  += __shfl_xor(rsx,  m, 32);
    rsx2 += __shfl_xor(rsx2, m, 32);
  }
  if (lane == 0) { wfit[wave] = fitl; wsx[wave] = rsx; wsx2[wave] = rsx2; }
  __syncthreads();

  if (tid == 0) {
    float F = 0.f, S = 0.f, S2 = 0.f;
#pragma unroll
    for (int w = 0; w < 8; ++w) { F += wfit[w]; S += wsx[w]; S2 += wsx2[w]; }
    pfit[blockIdx.x] = F; psx[blockIdx.x] = S; psx2[blockIdx.x] = S2;
  }

  // ---- Gather + write out[b, e, t] = codebook[best[t]][e] (t-coalesced) ----
  const size_t obase = (size_t)bblk * (EE * TT);
#pragma unroll 4
  for (int it = 0; it < 32; ++it) {
    int flat = (it << 8) + tid;              // 0..8191
    int e  = flat >> 7;
    int dt = flat & 127;
    out[obase + (size_t)e * TT + t0 + dt] = cb[(size_t)best_lds[dt] * EE + e];
  }
}

// ---------------------------------------------------------------------------
// Kernel 3: deterministic final reduction -> commit_loss, fit, x_norm
// ---------------------------------------------------------------------------
__global__ __launch_bounds__(256) void vq_finalize_kernel(const float* __restrict__ pfit,
                                                          const float* __restrict__ psx,
                                                          const float* __restrict__ psx2,
                                                          float* __restrict__ scalars) {
  __shared__ double rF[8], rS[8], rS2[8];
  const int tid = threadIdx.x, lane = tid & 31, wave = tid >> 5;
  double F = 0.0, S = 0.0, S2 = 0.0;
#pragma unroll
  for (int i = 0; i < NBLK / 256; ++i) {
    int idx = (i << 8) + tid;
    F += (double)pfit[idx]; S += (double)psx[idx]; S2 += (double)psx2[idx];
  }
#pragma unroll
  for (int m = 1; m <= 16; m <<= 1) {
    F  += __shfl_xor(F,  m, 32);
    S  += __shfl_xor(S,  m, 32);
    S2 += __shfl_xor(S2, m, 32);
  }
  if (lane == 0) { rF[wave] = F; rS[wave] = S; rS2[wave] = S2; }
  __syncthreads();
  if (tid == 0) {
    double f = 0.0, s = 0.0, s2 = 0.0;
#pragma unroll
    for (int w = 0; w < 8; ++w) { f += rF[w]; s += rS[w]; s2 += rS2[w]; }
    const double n_elem = (double)OUT_ELEMS;      // 8388608
    double fit    = f / (double)NROW;             // mean over rows
    double commit = f / n_elem;                   // sum((out-x)^2)/n
    double mean   = s / n_elem;
    double var    = s2 / n_elem - mean * mean;
    scalars[0] = (float)commit;
    scalars[1] = (float)fit;
    scalars[2] = (float)sqrt(var > 0.0 ? var : 0.0);
  }
}

// ---------------------------------------------------------------------------
extern "C" void kernel_launch(void* const* d_in, const int* in_sizes, int n_in,
                              void* d_out, int out_size, void* d_ws, size_t ws_size,
                              hipStream_t stream) {
  const float* x  = (const float*)d_in[0];   // [B,E,T] f32
  const float* cb = (const float*)d_in[1];   // [K,E]   f32
  float* out = (float*)d_out;                // [B,E,T] + 3 scalars

  float* ws   = (float*)d_ws;
  float* cb2  = ws;               // 2048
  float* pfit = ws + 2048;        // 1024
  float* psx  = ws + 3072;        // 1024
  float* psx2 = ws + 4096;        // 1024

  vq_cb2_kernel<<<KCB / 256, 256, 0, stream>>>(cb, cb2);
  vq_main_kernel<<<NBLK, 256, 0, stream>>>(x, cb, cb2, out, pfit, psx, psx2);
  vq_finalize_kernel<<<1, 256, 0, stream>>>(pfit, psx, psx2, out + OUT_ELEMS);
}